// EVI_Maxpool_19353122636474
// MI455X (gfx1250) — compile-verified
//
#include <hip/hip_runtime.h>
#include <stdint.h>

#define NB   16      // batch
#define WDIM 64      // H = W = 64
#define HW   4096    // 64*64 = N features
#define NP   1024    // pooled spatial size P = 32*32
#define PWD  32      // pooled width

// ---------------------------------------------------------------------------
// Kernel 1: 2x2 maxpool with flat argmax (first-max tie-break, torch order).
// mu: [16,1,64,64] -> mu_p: [16,1,32,32] (d_out head), ind: [16,1024] (ws)
// ---------------------------------------------------------------------------
__global__ __launch_bounds__(256)
void evi_maxpool_idx(const float* __restrict__ mu,
                     float* __restrict__ mu_p,
                     int* __restrict__ ind) {
    int idx = blockIdx.x * blockDim.x + threadIdx.x;   // 0 .. 16*1024-1
    if (idx >= NB * NP) return;
    int b  = idx >> 10;
    int p  = idx & (NP - 1);
    int ph = p >> 5;
    int pw = p & 31;
    const float* base = mu + (size_t)b * HW + (size_t)(2 * ph) * WDIM + 2 * pw;
    float v00 = base[0];
    float v01 = base[1];
    float v10 = base[WDIM];
    float v11 = base[WDIM + 1];
    // window order (kh,kw) row-major; strict '>' keeps the FIRST max
    float best = v00; int ki = 0;
    if (v01 > best) { best = v01; ki = 1; }
    if (v10 > best) { best = v10; ki = 2; }
    if (v11 > best) { best = v11; ki = 3; }
    int kh = ki >> 1, kw = ki & 1;
    mu_p[idx] = best;
    ind[idx]  = (2 * ph + kh) * WDIM + (2 * pw + kw);
}

// ---------------------------------------------------------------------------
// Kernel 2: sigma_p[b, r, c] = sigma[b, ind[b,c], ind[b,r]]
// 32x32 output tile per block, gathered asynchronously into LDS in the
// read-coalesced orientation (fixed sigma row per lane-group, gathered
// columns span ~256B), then stored fully coalesced.
// Uses CDNA5 async-tensor data path: global_load_async_to_lds_b32 (ASYNCcnt).
// ---------------------------------------------------------------------------
__global__ __launch_bounds__(256)
void evi_sigma_gather(const float* __restrict__ sigma,
                      const int* __restrict__ ind,
                      float* __restrict__ out) {
    __shared__ float tile[32][33];   // [r][c], pad -> conflict-free both phases
    __shared__ int   indr[32];
    __shared__ int   indc[32];

    const int tx = threadIdx.x;      // 0..31
    const int ty = threadIdx.y;      // 0..7
    const int ct = blockIdx.x;       // column tile
    const int rt = blockIdx.y;       // row tile
    const int b  = blockIdx.z;       // batch

    if (ty == 0)      indr[tx] = ind[b * NP + rt * 32 + tx];
    else if (ty == 1) indc[tx] = ind[b * NP + ct * 32 + tx];
    __syncthreads();

    const float* base = sigma + (size_t)b * HW * HW;   // batch slice (SGPR pair)

    // Phase 1: async gather  tile[r=tx][c] = sigma[b, indc[c], indr[tx]]
    const uint32_t roff = (uint32_t)indr[tx] * 4u;     // column byte offset
    #pragma unroll
    for (int k = 0; k < 4; ++k) {
        int c = ty + 8 * k;
        uint32_t goff  = (uint32_t)indc[c] * (uint32_t)(HW * 4) + roff; // < 2^31
        uint32_t laddr = (uint32_t)(uintptr_t)(void*)&tile[tx][c];      // AS3 offset
        asm volatile("global_load_async_to_lds_b32 %0, %1, %2"
                     :: "v"(laddr), "v"(goff), "s"(base) : "memory");
    }
    asm volatile("s_wait_asynccnt 0" ::: "memory");  // this wave's gathers done
    __syncthreads();                                  // all waves' LDS visible

    // Phase 2: coalesced stores  out[b, r, c=tx]  (128B per wave-row)
    float* obase = out + (size_t)b * NP * NP + (size_t)(rt * 32) * NP + ct * 32;
    #pragma unroll
    for (int k = 0; k < 4; ++k) {
        int r = ty + 8 * k;
        obase[(size_t)r * NP + tx] = tile[r][tx];
    }
}

// ---------------------------------------------------------------------------
// Launch: d_in[0]=mu (16*4096 f32), d_in[1]=sigma (16*4096*4096 f32)
// d_out = [ mu_p : 16384 f32 | sigma_p : 16*1024*1024 f32 ]
// d_ws  = ind (16*1024 int)
// ---------------------------------------------------------------------------
extern "C" void kernel_launch(void* const* d_in, const int* in_sizes, int n_in,
                              void* d_out, int out_size, void* d_ws, size_t ws_size,
                              hipStream_t stream) {
    const float* mu    = (const float*)d_in[0];
    const float* sigma = (const float*)d_in[1];
    float* mu_p    = (float*)d_out;
    float* sigma_p = (float*)d_out + NB * NP;   // after the 16384 mu_p floats
    int*   ind     = (int*)d_ws;

    // Kernel 1: maxpool + indices (16K threads)
    dim3 blk1(256), grd1((NB * NP + 255) / 256);
    evi_maxpool_idx<<<grd1, blk1, 0, stream>>>(mu, mu_p, ind);

    // Kernel 2: gathered 1024x1024 tiles per batch
    dim3 blk2(32, 8);
    dim3 grd2(NP / 32, NP / 32, NB);   // 32 x 32 x 16 = 16384 blocks
    evi_sigma_gather<<<grd2, blk2, 0, stream>>>(sigma, ind, sigma_p);
}